// FusedAGGemmFused_45664092291682
// MI455X (gfx1250) — compile-verified
//
#include <hip/hip_runtime.h>
#include <stdint.h>

// Fused AllGather+GEMM:  C[M,N] = sum_r  A_r[M,Kl] @ W[r*Kl:(r+1)*Kl, N]
//   gathered_act : [TP=8, M=8192, K_LOCAL=512] f32
//   local_W      : [K=4096, N_LOCAL=256]       f32
//   out          : [M=8192, N_LOCAL=256]       f32
//
// Native-f32 WMMA (V_WMMA_F32_16X16X4_F32) — exact fp32 math. ~17.2 GFLOP vs
// ~140 MB of traffic (entire working set fits the 192 MB L2), so the kernel is
// bound by the f32 matrix pipe. Data movement uses CDNA5 async Global->LDS
// (GLOBAL_LOAD_ASYNC_TO_LDS_B128/B32, ASYNCcnt) — no VGPR staging, and the
// W-tile transpose into [n][k] LDS layout comes free via per-lane LDS scatter.

typedef __attribute__((ext_vector_type(2))) float v2f;
typedef __attribute__((ext_vector_type(8))) float v8f;

#define MDIM 8192
#define KDIM 4096
#define NDIM 256
#define KLOC 512        // K per TP shard

#define BM 128
#define BN 64
#define BK 32
#define LDA 36          // BK + 4 pad: 16B-aligned b128 async stores, spreads banks
#define LDB 36
#define NKT (KDIM / BK) // 128 K-tiles; BK divides KLOC so a tile never crosses shards

__global__ __launch_bounds__(256)
void FusedAGGemm_wmma_async_kernel(const float* __restrict__ gact,
                                   const float* __restrict__ W,
                                   float* __restrict__ out) {
  __shared__ float sA[2][BM * LDA];   // A tile, row-major [m][k]
  __shared__ float sB[2][BN * LDB];   // W tile, TRANSPOSED [n][k]

  const int tid    = threadIdx.x;
  const int lane   = tid & 31;        // wave32
  const int wave   = tid >> 5;        // 8 waves per WG
  const int wm     = wave & 3;        // 4 waves along M
  const int wn     = wave >> 2;       // 2 waves along N
  const int blockM = blockIdx.y * BM;
  const int blockN = blockIdx.x * BN;

  // WMMA f32 16x16x4 fragment addressing (ISA 7.12.2):
  //   A: lane L holds row (L&15), K = kb0, kb0+1 where kb0 = 2*(L>=16)
  //   B: lane L holds col (L&15), same K split
  const int lrow = lane & 15;
  const int kb0  = (lane >> 4) << 1;  // 0 or 2

  v8f acc00 = {}, acc01 = {}, acc10 = {}, acc11 = {};

  // ---- K-tile-invariant per-lane addresses for the async copies ----
  // A tile: 128x32 f32 = 16 KB -> 4 x async b128 per thread-block (16B/lane).
  uint32_t aGOff[4], aLds0[4];
#pragma unroll
  for (int i = 0; i < 4; ++i) {
    int idx = tid + i * 256;              // 0..1023
    int row = idx >> 3;                   // 0..127
    int kq  = idx & 7;                    // b128 slot within BK
    aGOff[i] = (uint32_t)((row * KLOC + kq * 4) * 4);               // global byte off
    aLds0[i] = (uint32_t)(uintptr_t)&sA[0][row * LDA + kq * 4];     // LDS byte addr
  }
  // B tile: 32x64 f32 = 8 KB -> 8 x async b32 per thread-block (scatter-transpose).
  uint32_t bGOff[8], bLds0[8];
#pragma unroll
  for (int i = 0; i < 8; ++i) {
    int idx = tid + i * 256;              // 0..2047
    int kk  = idx >> 6;                   // 0..31
    int nn  = idx & 63;                   // coalesced along N in global
    bGOff[i] = (uint32_t)((kk * NDIM + nn) * 4);
    bLds0[i] = (uint32_t)(uintptr_t)&sB[0][nn * LDB + kk];
  }
  const uint32_t aBufStride = (uint32_t)(BM * LDA * 4);
  const uint32_t bBufStride = (uint32_t)(BN * LDB * 4);

  auto issueAsync = [&](int kt, int buf) {
    const int k0 = kt * BK;
    // shard r = k0/512; rows of that shard are contiguous with stride KLOC
    const uint64_t gA = (uint64_t)(uintptr_t)(gact +
        ((size_t)(k0 >> 9) * MDIM + blockM) * KLOC + (k0 & (KLOC - 1)));
    const uint64_t gB = (uint64_t)(uintptr_t)(W + (size_t)k0 * NDIM + blockN);
    const uint32_t aSel = buf ? aBufStride : 0u;
    const uint32_t bSel = buf ? bBufStride : 0u;
#pragma unroll
    for (int i = 0; i < 4; ++i)
      asm volatile("global_load_async_to_lds_b128 %0, %1, %2"
                   :: "v"(aLds0[i] + aSel), "v"(aGOff[i]), "s"(gA) : "memory");
#pragma unroll
    for (int i = 0; i < 8; ++i)
      asm volatile("global_load_async_to_lds_b32 %0, %1, %2"
                   :: "v"(bLds0[i] + bSel), "v"(bGOff[i]), "s"(gB) : "memory");
  };

  issueAsync(0, 0);

  for (int kt = 0; kt < NKT; ++kt) {
    // My asyncs (filling buf[kt&1]) done -> barrier -> everyone's done.
    asm volatile("s_wait_asynccnt 0x0" ::: "memory");
    __syncthreads();
    // buf[(kt+1)&1] is free (its last readers finished before the barrier).
    if (kt + 1 < NKT) issueAsync(kt + 1, (kt + 1) & 1);

    const int buf = kt & 1;
    const float* tA = &sA[buf][(wm * 32) * LDA];
    const float* tB = &sB[buf][(wn * 32) * LDB];
#pragma unroll
    for (int ks = 0; ks < 8; ++ks) {       // 8 k-steps of K=4
      const int kb = ks * 4 + kb0;
      v2f a0 = *(const v2f*)&tA[lrow * LDA + kb];
      v2f a1 = *(const v2f*)&tA[(lrow + 16) * LDA + kb];
      v2f b0 = *(const v2f*)&tB[lrow * LDB + kb];
      v2f b1 = *(const v2f*)&tB[(lrow + 16) * LDB + kb];
      acc00 = __builtin_amdgcn_wmma_f32_16x16x4_f32(false, a0, false, b0, (short)0, acc00, false, false);
      acc01 = __builtin_amdgcn_wmma_f32_16x16x4_f32(false, a0, false, b1, (short)0, acc01, false, false);
      acc10 = __builtin_amdgcn_wmma_f32_16x16x4_f32(false, a1, false, b0, (short)0, acc10, false, false);
      acc11 = __builtin_amdgcn_wmma_f32_16x16x4_f32(false, a1, false, b1, (short)0, acc11, false, false);
    }
  }

  // Epilogue: C/D layout — VGPR v holds row v (lanes 0-15) / row v+8 (lanes 16-31)
  const int crow = (lane >> 4) << 3;       // 0 or 8
  const int ccol = lane & 15;
  float* gC = out + (size_t)(blockM + wm * 32 + crow) * NDIM + (blockN + wn * 32 + ccol);
  v8f accs[4] = {acc00, acc01, acc10, acc11};
#pragma unroll
  for (int mi = 0; mi < 2; ++mi) {
#pragma unroll
    for (int ni = 0; ni < 2; ++ni) {
      v8f c = accs[mi * 2 + ni];
      float* p = gC + (size_t)(mi * 16) * NDIM + ni * 16;
#pragma unroll
      for (int v = 0; v < 8; ++v) p[(size_t)v * NDIM] = c[v];
    }
  }
}

extern "C" void kernel_launch(void* const* d_in, const int* in_sizes, int n_in,
                              void* d_out, int out_size, void* d_ws, size_t ws_size,
                              hipStream_t stream) {
  (void)in_sizes; (void)n_in; (void)out_size; (void)d_ws; (void)ws_size;
  const float* gact = (const float*)d_in[0];  // [8, 8192, 512] f32
  const float* W    = (const float*)d_in[1];  // [4096, 256]    f32
  float* out        = (float*)d_out;          // [8192, 256]    f32

  dim3 grid(NDIM / BN, MDIM / BM);  // (4, 64) = 256 workgroups x 8 waves
  dim3 block(256, 1, 1);
  FusedAGGemm_wmma_async_kernel<<<grid, block, 0, stream>>>(gact, W, out);
}